// PatientNeuralODE_9655086482040
// MI455X (gfx1250) — compile-verified
//
#include <hip/hip_runtime.h>

typedef float v2f __attribute__((ext_vector_type(2)));
typedef float v8f __attribute__((ext_vector_type(8)));

namespace {
constexpr int kBatch = 16384;
constexpr int kDin = 32;
constexpr int kDout = 16;
constexpr int kH = 128;
constexpr int kS = 48;
constexpr int kT = 64;
constexpr int kNSub = 8;

// LDS row strides (dwords). All ≡ 4 (mod 8): 8B-aligned b64 fragment loads,
// and n -> (stride*n) mod 64 is injective over 16 rows => conflict-free banks.
constexpr int kW1s = 52;   // W1^T  [128][52]  (48 used)
constexpr int kW2s = 132;  // W2^T  [128][132] (128 used)
constexpr int kW3s = 132;  // W3^T  [48][132]  (128 used)
constexpr int kWes = 36;   // Wenc^T[16][36]   (32 used)
constexpr int kWos = 52;   // Wout^T[16][52]   (48 used)
constexpr int kXs  = 132;  // per-wave activation staging [16][132]

constexpr int kWaves = 8;  // waves per block
constexpr int kRows  = 16; // batch rows per wave
}

__device__ __forceinline__ v8f wmma_f32(v2f a, v2f b, v8f c) {
  // D = A(16x4,f32) * B(4x16,f32) + C(16x16,f32)
  return __builtin_amdgcn_wmma_f32_16x16x4_f32(false, a, false, b, (short)0, c,
                                               false, false);
}

template <int KF>
__device__ __forceinline__ void load_a(v2f* a, const float* sXw, int aOff) {
#pragma unroll
  for (int kk = 0; kk < KF; ++kk)
    a[kk] = *(const v2f*)(sXw + aOff + kk * 4);
}

template <int KF>
__device__ __forceinline__ v8f tile_mm(const v2f* a, const float* wT,
                                       int rowOff, float bias) {
  v8f c;
#pragma unroll
  for (int v = 0; v < 8; ++v) c[v] = bias;
#pragma unroll
  for (int kk = 0; kk < KF; ++kk) {
    v2f b = *(const v2f*)(wT + rowOff + kk * 4);
    c = wmma_f32(a[kk], b, c);
  }
  return c;
}

__device__ __forceinline__ void stage_tile(float* sXw, int stBase, int j, v8f c) {
#pragma unroll
  for (int v = 0; v < 8; ++v)
    sXw[stBase + v * kXs + j * 16] = c[v];
}

__device__ __forceinline__ v8f tanh8(v8f x) {
  v8f r;
#pragma unroll
  for (int v = 0; v < 8; ++v) {
    float xv = x[v];
    float e = __expf(-2.0f * fabsf(xv));      // hw v_exp
    float t = __fdividef(1.0f - e, 1.0f + e); // stable tanh(|x|)
    r[v] = copysignf(t, xv);
  }
  return r;
}

struct Ctx {
  float* sXw;
  const float* sW1t;
  const float* sW2t;
  const float* sW3t;
  const float* sB1;
  const float* sB2;
  const float* sB3;
  int laneM, kHalf, stBase, aOff;
};

// k = f(xs) = W3^T tanh(W2^T tanh(W1^T xs + b1) + b2) + b3   (per 16-row tile)
__device__ __forceinline__ void eval_f(const Ctx& c, const v8f xs[3], v8f k[3]) {
#pragma unroll
  for (int j = 0; j < 3; ++j) stage_tile(c.sXw, c.stBase, j, xs[j]);
  v2f a[32];
  load_a<12>(a, c.sXw, c.aOff);  // A frags for K=48 kept in regs
#pragma unroll 1
  for (int nn = 0; nn < 8; ++nn) {
    v8f h = tile_mm<12>(a, c.sW1t, (nn * 16 + c.laneM) * kW1s + c.kHalf,
                        c.sB1[nn * 16 + c.laneM]);
    stage_tile(c.sXw, c.stBase, nn, tanh8(h));
  }
  load_a<32>(a, c.sXw, c.aOff);  // K=128
#pragma unroll 1
  for (int nn = 0; nn < 8; ++nn) {
    v8f h = tile_mm<32>(a, c.sW2t, (nn * 16 + c.laneM) * kW2s + c.kHalf,
                        c.sB2[nn * 16 + c.laneM]);
    stage_tile(c.sXw, c.stBase, nn, tanh8(h));
  }
  load_a<32>(a, c.sXw, c.aOff);
#pragma unroll
  for (int nn = 0; nn < 3; ++nn) {
    k[nn] = tile_mm<32>(a, c.sW3t, (nn * 16 + c.laneM) * kW3s + c.kHalf,
                        c.sB3[nn * 16 + c.laneM]);
  }
}

__device__ __forceinline__ void write_step(const Ctx& c, const float* sWot,
                                           const float* sBo, const v8f x[3],
                                           int t, int row0, int lane, int laneHi,
                                           float* __restrict__ outp,
                                           float* __restrict__ states) {
#pragma unroll
  for (int j = 0; j < 3; ++j) stage_tile(c.sXw, c.stBase, j, x[j]);
  // states[b][t][s]
#pragma unroll
  for (int i = 0; i < 24; ++i) {
    int d = i * 32 + lane;
    int m = d / kS;
    int n = d - m * kS;
    states[((size_t)(row0 + m) * kT + t) * kS + n] = c.sXw[m * kXs + n];
  }
  // outputs[b][t][o] = sigmoid(x @ W_out + b_out)
  v2f a[12];
  load_a<12>(a, c.sXw, c.aOff);
  v8f o = tile_mm<12>(a, sWot, c.laneM * kWos + c.kHalf, sBo[c.laneM]);
#pragma unroll
  for (int v = 0; v < 8; ++v) {
    float sv = __fdividef(1.0f, 1.0f + __expf(-o[v]));
    int m = laneHi * 8 + v;
    outp[((size_t)(row0 + m) * kT + t) * kDout + c.laneM] = sv;
  }
}

__global__ __launch_bounds__(kWaves * 32) void patient_node_wmma(
    const float* __restrict__ meas, const float* __restrict__ pts,
    const float* __restrict__ Wenc, const float* __restrict__ benc,
    const float* __restrict__ W1, const float* __restrict__ b1,
    const float* __restrict__ W2, const float* __restrict__ b2,
    const float* __restrict__ W3, const float* __restrict__ b3,
    const float* __restrict__ Wout, const float* __restrict__ bout,
    float* __restrict__ outp, float* __restrict__ states) {
  __shared__ float sW1t[kH * kW1s];
  __shared__ float sW2t[kH * kW2s];
  __shared__ float sW3t[kS * kW3s];
  __shared__ float sWet[kDout * kWes];
  __shared__ float sWot[kDout * kWos];
  __shared__ float sB1[kH], sB2[kH], sB3[kS], sBe[kDout], sBo[kDout];
  __shared__ float sX[kWaves * kRows * kXs];

  const int tid = threadIdx.x;
  const int nthr = kWaves * 32;
  // Stage all weights transposed into LDS once.
  for (int i = tid; i < kS * kH; i += nthr) {
    int s = i >> 7, h = i & 127;
    sW1t[h * kW1s + s] = W1[i];
  }
  for (int i = tid; i < kH * kH; i += nthr) {
    int k = i >> 7, n = i & 127;
    sW2t[n * kW2s + k] = W2[i];
  }
  for (int i = tid; i < kH * kS; i += nthr) {
    int h = i / kS, s = i - h * kS;
    sW3t[s * kW3s + h] = W3[i];
  }
  for (int i = tid; i < kDin * kDout; i += nthr) {
    int d = i >> 4, o = i & 15;
    sWet[o * kWes + d] = Wenc[i];
  }
  for (int i = tid; i < kS * kDout; i += nthr) {
    int s = i >> 4, o = i & 15;
    sWot[o * kWos + s] = Wout[i];
  }
  if (tid < kH) { sB1[tid] = b1[tid]; sB2[tid] = b2[tid]; }
  if (tid < kS) sB3[tid] = b3[tid];
  if (tid < kDout) { sBe[tid] = benc[tid]; sBo[tid] = bout[tid]; }
  __syncthreads();

  const int lane = tid & 31;
  const int wave = tid >> 5;
  const int laneM = lane & 15;
  const int laneHi = lane >> 4;

  Ctx c;
  c.sXw = sX + wave * (kRows * kXs);
  c.sW1t = sW1t; c.sW2t = sW2t; c.sW3t = sW3t;
  c.sB1 = sB1; c.sB2 = sB2; c.sB3 = sB3;
  c.laneM = laneM;
  c.kHalf = laneHi * 2;
  c.stBase = laneHi * 8 * kXs + laneM;
  c.aOff = laneM * kXs + c.kHalf;

  const int row0 = (blockIdx.x * kWaves + wave) * kRows;

  // x0 = [measurements | meas @ W_enc + b_enc], kept as 3 C-layout tiles.
  v8f x[3];
#pragma unroll
  for (int j = 0; j < 2; ++j) {
#pragma unroll
    for (int v = 0; v < 8; ++v) {
      int m = laneHi * 8 + v;
      x[j][v] = meas[(size_t)(row0 + m) * kDin + j * 16 + laneM];
    }
  }
  stage_tile(c.sXw, c.stBase, 0, x[0]);
  stage_tile(c.sXw, c.stBase, 1, x[1]);
  {
    v2f a[8];
    load_a<8>(a, c.sXw, c.aOff);
    x[2] = tile_mm<8>(a, sWet, laneM * kWes + c.kHalf, sBe[laneM]);
  }

  write_step(c, sWot, sBo, x, 0, row0, lane, laneHi, outp, states);

  // Dopri5 coefficients
  constexpr float B21 = 1.0f / 5.0f;
  constexpr float B31 = 3.0f / 40.0f, B32 = 9.0f / 40.0f;
  constexpr float B41 = 44.0f / 45.0f, B42 = -56.0f / 15.0f, B43 = 32.0f / 9.0f;
  constexpr float B51 = 19372.0f / 6561.0f, B52 = -25360.0f / 2187.0f,
                  B53 = 64448.0f / 6561.0f, B54 = -212.0f / 729.0f;
  constexpr float B61 = 9017.0f / 3168.0f, B62 = -355.0f / 33.0f,
                  B63 = 46732.0f / 5247.0f, B64 = 49.0f / 176.0f,
                  B65 = -5103.0f / 18656.0f;
  constexpr float C1 = 35.0f / 384.0f, C3 = 500.0f / 1113.0f,
                  C4 = 125.0f / 192.0f, C5 = -2187.0f / 6784.0f,
                  C6 = 11.0f / 84.0f;

  float tprev = pts[0];
#pragma unroll 1
  for (int t = 1; t < kT; ++t) {
    float tcur = pts[t];
    float hs = (tcur - tprev) * (1.0f / kNSub);
    tprev = tcur;
#pragma unroll 1
    for (int ss = 0; ss < kNSub; ++ss) {
      v8f k1[3], k2[3], k3[3], k4[3], k5[3], k6[3], xs[3];
      eval_f(c, x, k1);
#pragma unroll
      for (int j = 0; j < 3; ++j)
#pragma unroll
        for (int v = 0; v < 8; ++v)
          xs[j][v] = x[j][v] + hs * (B21 * k1[j][v]);
      eval_f(c, xs, k2);
#pragma unroll
      for (int j = 0; j < 3; ++j)
#pragma unroll
        for (int v = 0; v < 8; ++v)
          xs[j][v] = x[j][v] + hs * (B31 * k1[j][v] + B32 * k2[j][v]);
      eval_f(c, xs, k3);
#pragma unroll
      for (int j = 0; j < 3; ++j)
#pragma unroll
        for (int v = 0; v < 8; ++v)
          xs[j][v] = x[j][v] +
                     hs * (B41 * k1[j][v] + B42 * k2[j][v] + B43 * k3[j][v]);
      eval_f(c, xs, k4);
#pragma unroll
      for (int j = 0; j < 3; ++j)
#pragma unroll
        for (int v = 0; v < 8; ++v)
          xs[j][v] = x[j][v] + hs * (B51 * k1[j][v] + B52 * k2[j][v] +
                                     B53 * k3[j][v] + B54 * k4[j][v]);
      eval_f(c, xs, k5);
#pragma unroll
      for (int j = 0; j < 3; ++j)
#pragma unroll
        for (int v = 0; v < 8; ++v)
          xs[j][v] = x[j][v] + hs * (B61 * k1[j][v] + B62 * k2[j][v] +
                                     B63 * k3[j][v] + B64 * k4[j][v] +
                                     B65 * k5[j][v]);
      eval_f(c, xs, k6);
#pragma unroll
      for (int j = 0; j < 3; ++j)
#pragma unroll
        for (int v = 0; v < 8; ++v)
          x[j][v] = x[j][v] + hs * (C1 * k1[j][v] + C3 * k3[j][v] +
                                    C4 * k4[j][v] + C5 * k5[j][v] +
                                    C6 * k6[j][v]);
    }
    write_step(c, sWot, sBo, x, t, row0, lane, laneHi, outp, states);
  }
}

extern "C" void kernel_launch(void* const* d_in, const int* in_sizes, int n_in,
                              void* d_out, int out_size, void* d_ws,
                              size_t ws_size, hipStream_t stream) {
  (void)in_sizes; (void)n_in; (void)out_size; (void)d_ws; (void)ws_size;
  const float* meas = (const float*)d_in[0];
  const float* pts  = (const float*)d_in[1];
  const float* Wenc = (const float*)d_in[2];
  const float* benc = (const float*)d_in[3];
  const float* W1   = (const float*)d_in[4];
  const float* b1   = (const float*)d_in[5];
  const float* W2   = (const float*)d_in[6];
  const float* b2   = (const float*)d_in[7];
  const float* W3   = (const float*)d_in[8];
  const float* b3   = (const float*)d_in[9];
  const float* Wout = (const float*)d_in[10];
  const float* bout = (const float*)d_in[11];
  float* outp = (float*)d_out;                              // [B][T][DOUT]
  float* states = outp + (size_t)kBatch * kT * kDout;       // [B][T][S]

  dim3 grid(kBatch / (kRows * kWaves));  // 128 blocks
  dim3 block(kWaves * 32);               // 256 threads = 8 waves
  hipLaunchKernelGGL(patient_node_wmma, grid, block, 0, stream, meas, pts,
                     Wenc, benc, W1, b1, W2, b2, W3, b3, Wout, bout, outp,
                     states);
}